// RoiCropper_79542794322056
// MI455X (gfx1250) — compile-verified
//
#include <hip/hip_runtime.h>
#include <stdint.h>

// ROI cropper: out[b,n] = image[b, z0:z0+32, y0:y0+32, x0:x0+32, 0]
// image: [8,128,128,128,1] f32 (64 MiB, L2-resident), boxes: [8,128,3] i32,
// out: [8,128,32,32,32,1] f32 (128 MiB). Pure data movement -> CDNA5 Tensor
// Data Mover: 3-D tiled DMA global->LDS, 1-D DMA LDS->global (NT stores).

typedef unsigned int v4u __attribute__((ext_vector_type(4)));
typedef int          v8i __attribute__((ext_vector_type(8)));
typedef int          v4i __attribute__((ext_vector_type(4)));

#define CHUNK_Z      8                       // z-planes per TDM sub-tile
#define CHUNK_FLOATS (32 * 32 * CHUNK_Z)     // 8192 floats = 32 KiB
#define CHUNK_BYTES  (CHUNK_FLOATS * 4)
#define NCHUNK       (32 / CHUNK_Z)          // 4 sub-tiles per crop

// cpol: gfx12-style CPol immediate, TH in bits [2:0]; 0 = RT, 1 = NT.
#define CPOL_RT 0
#define CPOL_NT 1

__device__ __forceinline__ void tdm_load(v4u g0, v8i g1, v4i g2, v4i g3) {
#if defined(__clang_major__) && (__clang_major__ >= 23)
    v8i z8 = {0, 0, 0, 0, 0, 0, 0, 0};
    __builtin_amdgcn_tensor_load_to_lds(g0, g1, g2, g3, z8, CPOL_RT);
#else
    __builtin_amdgcn_tensor_load_to_lds(g0, g1, g2, g3, CPOL_RT);
#endif
}

__device__ __forceinline__ void tdm_store(v4u g0, v8i g1, v4i g2, v4i g3) {
#if defined(__clang_major__) && (__clang_major__ >= 23)
    v8i z8 = {0, 0, 0, 0, 0, 0, 0, 0};
    __builtin_amdgcn_tensor_store_from_lds(g0, g1, g2, g3, z8, CPOL_NT);
#else
    __builtin_amdgcn_tensor_store_from_lds(g0, g1, g2, g3, CPOL_NT);
#endif
}

__global__ __launch_bounds__(32)
void roi_crop_tdm_kernel(const float* __restrict__ image,
                         const int*   __restrict__ boxes,
                         float*       __restrict__ out)
{
    // double-buffered staging: 2 x 32 KiB (320 KiB/WGP available on CDNA5)
    __shared__ float lds_buf[2][CHUNK_FLOATS];

    const uint32_t crop = blockIdx.x;        // b*128 + n
    const uint32_t b    = crop >> 7;         // N == 128

    // box = (z0, y0, x0); force into SGPRs (descriptors live in SGPRs)
    int z0 = __builtin_amdgcn_readfirstlane(boxes[crop * 3 + 0]);
    int y0 = __builtin_amdgcn_readfirstlane(boxes[crop * 3 + 1]);
    int x0 = __builtin_amdgcn_readfirstlane(boxes[crop * 3 + 2]);

    const uint64_t img_base = (uint64_t)(uintptr_t)image;
    const uint64_t out_base = (uint64_t)(uintptr_t)out +
                              (uint64_t)crop * (uint64_t)(32 * 32 * 32 * 4);

    const uint32_t lds0 = (uint32_t)(uintptr_t)&lds_buf[0][0];
    const uint32_t lds1 = (uint32_t)(uintptr_t)&lds_buf[1][0];

    // ---- D# group1, LOAD: data_size=4B; tensor 128x128 (dim0/dim1),
    // tile 32(x) x 32(y) x CHUNK_Z(z); stride0=128, stride1=128*128=16384.
    const v8i g1_load = {
        0x00020000,              // data_size=2 (4B), mask=0, no pad/iterate
        0x00800000,              // tensor_dim0[15:0]=128   (bits 63:48)
        0x00800000,              // tensor_dim1[15:0]=128   (dword2[31:16])
        0x00200000,              // tile_dim0=32            (bits 127:112)
        (CHUNK_Z << 16) | 32,    // tile_dim1=32, tile_dim2=CHUNK_Z
        128,                     // tensor_dim0_stride[31:0] = W = 128
        0x40000000,              // tensor_dim1_stride[15:0] = 16384 (dword6[31:16])
        0
    };
    const v4i g2_load = {128, 0, 0, 0};      // tensor_dim2 = D = 128
    const v4i g_zero4 = {0, 0, 0, 0};

    // ---- D# group1, STORE: flat 1-D tile of CHUNK_FLOATS elements.
    const v8i g1_store = {
        0x00020000,              // data_size=4B
        CHUNK_FLOATS << 16,      // tensor_dim0[15:0] = 8192
        0,
        CHUNK_FLOATS << 16,      // tile_dim0 = 8192
        0,                       // tile_dim1 = tile_dim2 = 0 (1-D)
        CHUNK_FLOATS,            // tensor_dim0_stride = 8192
        0, 0
    };

    for (uint32_t k = 0; k < NCHUNK; ++k) {
        const uint32_t lds_off = (k & 1) ? lds1 : lds0;

        // global addr of tile start: image[b, z0 + k*CHUNK_Z, y0, x0]
        const uint64_t ga = img_base + 4ull *
            ((uint64_t)(b * 128u + (uint32_t)z0 + k * CHUNK_Z) * 16384ull +
             (uint32_t)y0 * 128ull + (uint32_t)x0);
        const uint64_t oa = out_base + (uint64_t)k * CHUNK_BYTES;

        // group0: count=1 | lds_addr | global_addr[56:0] | type=2 (bits 127:126)
        v4u g0l = { 1u, lds_off, (uint32_t)ga,
                    (uint32_t)((ga >> 32) & 0x01FFFFFFull) | 0x80000000u };
        tdm_load(g0l, g1_load, g2_load, g_zero4);

        // wait: sub-tile resident in LDS (also drains store k-1, so buffer
        // (k-1)&1 is free again before load k+1 overwrites it)
        __builtin_amdgcn_s_wait_tensorcnt(0);

        v4u g0s = { 1u, lds_off, (uint32_t)oa,
                    (uint32_t)((oa >> 32) & 0x01FFFFFFull) | 0x80000000u };
        tdm_store(g0s, g1_store, g_zero4, g_zero4);
        // no wait here: store k overlaps with load k+1 (other LDS buffer)
    }
    __builtin_amdgcn_s_wait_tensorcnt(0);
}

extern "C" void kernel_launch(void* const* d_in, const int* in_sizes, int n_in,
                              void* d_out, int out_size, void* d_ws, size_t ws_size,
                              hipStream_t stream) {
    const float* image = (const float*)d_in[0];
    const int*   boxes = (const int*)d_in[1];
    float*       out   = (float*)d_out;

    const int n_crops = in_sizes[1] / 3;     // B*N = 1024
    roi_crop_tdm_kernel<<<dim3(n_crops), dim3(32), 0, stream>>>(image, boxes, out);
}